// CapsuleLayer_14199161881177
// MI455X (gfx1250) — compile-verified
//
#include <hip/hip_runtime.h>
#include <math.h>

// Shapes from the reference: x is (B=64, NUM_UNITS=32, IN_CH=256, UNIT=128) fp32.
// After the reshape-reinterpret: u_hat[b,i,n,u] = x[b*1048576 + i*4096 + n*128 + u].
// Output: (64, 32, 128) fp32 = 262144 elements.

typedef __attribute__((ext_vector_type(2))) float v2f;
typedef __attribute__((ext_vector_type(4))) float v4f;
typedef __attribute__((ext_vector_type(8))) float v8f;

#define BATCH    64
#define NUNITS   32
#define INCH     256
#define UNIT     128
#define B_STRIDE (NUNITS * INCH * UNIT)   // 1048576 floats per batch
#define I_STRIDE (NUNITS * UNIT)          // 4096 floats per in-channel row

__global__ __launch_bounds__(256) void capsule_route_squash(
    const float* __restrict__ x, float* __restrict__ out) {
  // part[w][u]: per-wave partial column sums (stride 132 to de-conflict the
  // strided 16-dword reads in the WMMA staging phase: 2*132 % 64 != 0).
  __shared__ float part[8][132];
  __shared__ float sS[128];
  __shared__ float red[4];
  __shared__ float magS;

  const int tid  = threadIdx.x;
  const int w    = tid >> 5;       // wave id 0..7
  const int lane = tid & 31;       // wave32 lane
  const int bn   = blockIdx.x;     // 0..2047
  const int b    = bn >> 5;
  const int n    = bn & 31;

  // ---------------- Phase 1: streaming fp32 column reduction -------------
  // Wave w sums rows i = w, w+8, ..., w+248. Per load instruction a wave
  // covers one full contiguous 512B row (32 lanes x float4). Data has zero
  // reuse anywhere on chip -> non-temporal loads.
  const float* base = x + (size_t)b * B_STRIDE + (size_t)n * UNIT
                        + (size_t)w * I_STRIDE + (size_t)(lane << 2);
  v4f acc = {0.0f, 0.0f, 0.0f, 0.0f};
#pragma unroll 8
  for (int j = 0; j < 32; ++j) {
    v4f v = __builtin_nontemporal_load(
        (const v4f*)(base + (size_t)j * (8 * I_STRIDE)));
    acc += v;
  }
  *(v4f*)&part[w][lane << 2] = acc;   // 528-byte row stride keeps 16B align
  __syncthreads();

  // ---------------- Phase 2: 8 -> 1 combine on the fp32 matrix pipe ------
  // s[u] = (1/256) * sum_w part[w][u], done as two chained
  // V_WMMA_F32_16X16X4_F32: A(16x4) = partials (M = unit col, K = wave id),
  // B(4x16) = constant 1/256 (uniform, so B lane layout is irrelevant).
  // Wave w owns unit columns u0..u0+15.
  {
    const int half = lane >> 4;     // ISA A-layout: lanes 0-15 hold K=0,1; 16-31 hold K=2,3
    const int m    = lane & 15;     // M = u0 + m
    const int u0   = w << 4;

    v2f a0, a1;
    a0.x = part[2 * half + 0][u0 + m];
    a0.y = part[2 * half + 1][u0 + m];
    a1.x = part[4 + 2 * half + 0][u0 + m];
    a1.y = part[4 + 2 * half + 1][u0 + m];

    const v2f bones = {1.0f / 256.0f, 1.0f / 256.0f};
    v8f c = {};
    c = __builtin_amdgcn_wmma_f32_16x16x4_f32(false, a0, false, bones,
                                              (short)0, c, false, false);
    c = __builtin_amdgcn_wmma_f32_16x16x4_f32(false, a1, false, bones,
                                              (short)0, c, false, false);

    // D layout: lane l, vgpr v -> D[M = v + 8*(l/16)][N = l%16]; all N equal.
    if (m == 0) {
      const int ub = u0 + (half << 3);
#pragma unroll
      for (int v = 0; v < 8; ++v) sS[ub + v] = c[v];
    }
  }
  __syncthreads();

  // ---------------- Phase 3: squash over the 128-wide unit dim -----------
  if (tid < 128) {
    float s  = sS[tid];
    float ss = s * s;
#pragma unroll
    for (int off = 16; off > 0; off >>= 1) ss += __shfl_xor(ss, off, 32);
    if ((tid & 31) == 0) red[tid >> 5] = ss;
  }
  __syncthreads();
  if (tid == 0) magS = red[0] + red[1] + red[2] + red[3];
  __syncthreads();
  if (tid < 128) {
    const float ms    = magS;
    const float scale = (ms / (1.0f + ms)) / (sqrtf(ms) + 1e-5f);
    out[(size_t)bn * UNIT + tid] = scale * sS[tid];
  }
}

extern "C" void kernel_launch(void* const* d_in, const int* in_sizes, int n_in,
                              void* d_out, int out_size, void* d_ws, size_t ws_size,
                              hipStream_t stream) {
  (void)in_sizes; (void)n_in; (void)out_size; (void)d_ws; (void)ws_size;
  const float* x = (const float*)d_in[0];
  float* out     = (float*)d_out;
  capsule_route_squash<<<dim3(BATCH * NUNITS), dim3(256), 0, stream>>>(x, out);
}